// SemanticGuidedGate_68985764708611
// MI455X (gfx1250) — compile-verified
//
#include <hip/hip_runtime.h>
#include <math.h>

// Problem constants (match reference).
#define B_   4
#define K_   16
#define MS_  4096
#define NT_  16384

typedef __attribute__((ext_vector_type(2))) float v2f;
typedef __attribute__((ext_vector_type(8))) float v8f;

// ---------------------------------------------------------------------------
// Kernel A: collapse softmax+1x1conv to a scalar g[b,m], and pack sources as
// float4 (sx, sy, sz, -0.5*|s|^2) for the WMMA B-fragment.
// One thread per (b, m): B*MS = 16384 threads.
// ---------------------------------------------------------------------------
__global__ void sgg_precompute(const float* __restrict__ sem,   // [B,K,MS]
                               const float* __restrict__ spos,  // [B,3,MS]
                               const float* __restrict__ convw, // [K]
                               const float* __restrict__ convb, // [1]
                               float4* __restrict__ S4,         // [B*MS]
                               float* __restrict__ g)           // [B*MS]
{
    const int tid = blockIdx.x * blockDim.x + threadIdx.x;   // 0 .. B*MS-1
    const int b   = tid >> 12;          // / MS_ (4096)
    const int m   = tid & (MS_ - 1);

    // Softmax over K (stride MS in k), fused with conv weight dot product.
    const float* lp = sem + ((size_t)b * K_) * MS_ + m;
    float mx = -3.402823466e38f;
#pragma unroll
    for (int k = 0; k < K_; ++k) mx = fmaxf(mx, lp[(size_t)k * MS_]);
    float se = 0.f, sw = 0.f;
#pragma unroll
    for (int k = 0; k < K_; ++k) {
        float e = __expf(lp[(size_t)k * MS_] - mx);
        se += e;
        sw += convw[k] * e;
    }
    g[tid] = sw / se + convb[0];

    // Pack source point for the 16x16x4 WMMA B matrix.
    const float sx = spos[((size_t)b * 3 + 0) * MS_ + m];
    const float sy = spos[((size_t)b * 3 + 1) * MS_ + m];
    const float sz = spos[((size_t)b * 3 + 2) * MS_ + m];
    S4[tid] = make_float4(sx, sy, sz, -0.5f * (sx * sx + sy * sy + sz * sz));
}

// ---------------------------------------------------------------------------
// Kernel B: nearest-neighbor argmax via V_WMMA_F32_16X16X4_F32.
//   score[t,s] = t.x*s.x + t.y*s.y + t.z*s.z + 1*(-0.5|s|^2)
//   argmin ||t-s||^2  ==  argmax score[t,s]
// Each wave32 owns one 16-target tile; loops over MS/16 = 256 source blocks,
// unrolled x2 so two independent load->WMMA->argmax chains overlap the
// WMMA->VALU hazard window.
// C/D layout: VGPR r holds target m=r (lanes 0-15) / m=r+8 (lanes 16-31),
// source column n = blk*16 + (lane & 15).
// ---------------------------------------------------------------------------
__global__ void sgg_nn_gate(const float* __restrict__ tpos, // [B,3,NT]
                            const v2f*  __restrict__ S4v,   // [B*MS] as 2x v2f
                            const float* __restrict__ g,    // [B*MS]
                            float* __restrict__ out)        // [B*NT]
{
    const int lane = threadIdx.x & 31;
    const int wave = threadIdx.x >> 5;
    const int tile = blockIdx.x * 8 + wave;     // 0 .. 4095 (exact)
    const int b     = tile >> 10;               // / (NT/16 = 1024)
    const int tBase = (tile & 1023) << 4;       // target tile base
    const int ln = lane & 15;
    const int hi = lane >> 4;

    // --- Build A fragment (16 targets x 4, f32): lane<16 -> K=0,1; lane>=16 -> K=2,3
    const int   nt = tBase + ln;
    const float* tp = tpos + (size_t)b * 3 * NT_ + nt;
    const float tx = tp[0];
    const float ty = tp[NT_];
    const float tz = tp[2 * NT_];
    v2f a;
    a.x = hi ? tz : tx;   // K=0 or K=2
    a.y = hi ? 1.f : ty;  // K=1 or K=3  (K=3 element == 1.0 picks up -0.5|s|^2)

    float bestV[8];
    int   bestI[8];
#pragma unroll
    for (int r = 0; r < 8; ++r) { bestV[r] = -3.402823466e38f; bestI[r] = 0; }

    // Per-lane B-fragment source: float2 at (base+ln) with low/high half by
    // lane group. S4 viewed as v2f[]: source n -> elements 2n (x,y), 2n+1 (z,w).
    const v2f* sp = S4v + (size_t)b * MS_ * 2 + (size_t)ln * 2 + hi;

    for (int blk = 0; blk < MS_ / 16; blk += 2) {
        v2f bf0 = sp[0];
        v2f bf1 = sp[32];
        sp += 64;                       // advance 32 sources (64 v2f)

        v8f c0 = {};
        c0 = __builtin_amdgcn_wmma_f32_16x16x4_f32(
                false, a, false, bf0, (short)0, c0, false, false);
        v8f c1 = {};
        c1 = __builtin_amdgcn_wmma_f32_16x16x4_f32(
                false, a, false, bf1, (short)0, c1, false, false);

        const int n0 = (blk << 4) + ln;
        const int n1 = n0 + 16;
#pragma unroll
        for (int r = 0; r < 8; ++r) {
            if (c0[r] > bestV[r]) { bestV[r] = c0[r]; bestI[r] = n0; }
        }
#pragma unroll
        for (int r = 0; r < 8; ++r) {
            if (c1[r] > bestV[r]) { bestV[r] = c1[r]; bestI[r] = n1; }
        }
    }

    // --- Cross-lane argmax over the 16-lane N groups (masks 1,2,4,8 stay
    // in-half). Branchless: non-short-circuit condition + selects so this
    // lowers to v_cmp/s_and/v_cndmask instead of exec-mask branches.
#pragma unroll
    for (int r = 0; r < 8; ++r) {
        float v = bestV[r];
        int   i = bestI[r];
#pragma unroll
        for (int m = 8; m >= 1; m >>= 1) {
            const float ov = __shfl_xor(v, m, 32);
            const int   oi = __shfl_xor(i, m, 32);
            const int take = (int)(ov > v) | ((int)(ov == v) & (int)(oi < i));
            v = take ? ov : v;
            i = take ? oi : i;
        }
        bestV[r] = v;
        bestI[r] = i;
    }

    // lanes 0 / 16 hold the result for targets m = r / r + 8.
    if (ln == 0) {
        const int mbase = hi * 8;
        const float* gb = g + (size_t)b * MS_;
        float* ob = out + (size_t)b * NT_ + tBase + mbase;
#pragma unroll
        for (int r = 0; r < 8; ++r) {
            const float gv = gb[bestI[r]];
            ob[r] = 1.f / (1.f + __expf(-gv));
        }
    }
}

// ---------------------------------------------------------------------------
extern "C" void kernel_launch(void* const* d_in, const int* in_sizes, int n_in,
                              void* d_out, int out_size, void* d_ws, size_t ws_size,
                              hipStream_t stream) {
    const float* sem   = (const float*)d_in[0];  // [B,K,MS]
    const float* spos  = (const float*)d_in[1];  // [B,3,MS]
    const float* tpos  = (const float*)d_in[2];  // [B,3,NT]
    const float* convw = (const float*)d_in[3];  // [1,K,1] -> K floats
    const float* convb = (const float*)d_in[4];  // [1]

    // Workspace layout: S4 (B*MS float4 = 256KB) then g (B*MS float = 64KB).
    float4* S4 = (float4*)d_ws;
    float*  g  = (float*)((char*)d_ws + (size_t)B_ * MS_ * sizeof(float4));

    // Kernel A: 16384 threads.
    sgg_precompute<<<(B_ * MS_) / 256, 256, 0, stream>>>(sem, spos, convw, convb, S4, g);

    // Kernel B: 4096 wave-tiles, 8 waves per 256-thread block -> 512 blocks.
    sgg_nn_gate<<<(B_ * (NT_ / 16)) / 8, 256, 0, stream>>>(
        tpos, (const v2f*)S4, g, (float*)d_out);
}